// _JambaSSMCore_65687229825529
// MI455X (gfx1250) — compile-verified
//
#include <hip/hip_runtime.h>
#include <math.h>

typedef __attribute__((ext_vector_type(2))) float v2f;
typedef __attribute__((ext_vector_type(8))) float v8f;

// ---- problem constants ----
#define BB 2
#define TT 2048
#define DM 1024
#define DS 16
#define NPROJ (DM + 2 * DS)   // 1056
#define MM (BB * TT)          // 4096
#define KK DM                 // 1024
#define NC 32                 // chunks
#define CHT (TT / NC)         // 64 steps / chunk

// ws layout (floats)
#define PROJ_ELEMS (MM * NPROJ)            // 4,325,376
#define PQ_ELEMS (BB * DM * NC * DS)       // 1,048,576
#define Y_ELEMS (BB * TT * DM)             // 4,194,304

__device__ __forceinline__ float softplus_f(float z) {
    // numerically stable softplus: log(1 + e^z)
    return (z > 20.0f) ? z : log1pf(__expf(z));
}

// ---------------------------------------------------------------------------
// Kernel 1: proj = x2d(4096x1024) @ W(1024x1056), fp32 WMMA 16x16x4.
// Block: 256 thr = 8 waves covering 32(M) x 256(N). Each wave owns a 2x2
// grid of 16x16 tiles (register reuse: 2 A-frags + 2 B-frags -> 4 WMMAs per
// k-step). A/W chunks staged through LDS, K chunked at 32 (36 KB LDS).
// Out-of-range N columns are zero-filled in LDS; only stores are guarded,
// so EXEC stays all-1s through every WMMA.
// ---------------------------------------------------------------------------
#define BM 32
#define BN 256
#define KC 32

__global__ __launch_bounds__(256) void gemm_proj_kernel(
    const float* __restrict__ X, const float* __restrict__ W,
    float* __restrict__ P) {
    __shared__ float As[BM][KC];     // 4 KB
    __shared__ float Ws[KC][BN];     // 32 KB

    const int t = threadIdx.x;
    const int wave = t >> 5;
    const int lane = t & 31;
    const int m0 = blockIdx.x * BM;
    const int n0 = blockIdx.y * BN;

    v8f acc00 = {}, acc01 = {}, acc10 = {}, acc11 = {};

    const int r = lane & 15;           // A row within tile / D col within tile
    const int kh = (lane >> 4) << 1;   // K half select: 0 or 2
    const int col0 = (wave << 5) + (lane & 15);   // wave's first N tile col
    const int col1 = col0 + 16;                   // wave's second N tile col

    for (int kb = 0; kb < KK; kb += KC) {
        // --- stage A: 32x32 floats, one float4 per thread ---
        {
            int ra = t >> 3;
            int ca = (t & 7) << 2;
            *(float4*)&As[ra][ca] =
                *(const float4*)&X[(m0 + ra) * KK + kb + ca];
        }
        // --- stage W: 32x256 floats, 8 float4 per thread, N-guarded ---
#pragma unroll
        for (int i = 0; i < 8; ++i) {
            int idx = t + (i << 8);
            int kr = idx >> 6;
            int cw = (idx & 63) << 2;
            int n = n0 + cw;
            float4 v = make_float4(0.f, 0.f, 0.f, 0.f);
            if (n < NPROJ) v = *(const float4*)&W[(kb + kr) * NPROJ + n];
            *(float4*)&Ws[kr][cw] = v;
        }
        __syncthreads();

#pragma unroll
        for (int kk = 0; kk < KC; kk += 4) {
            v2f a0, a1, b0, b1;
            a0.x = As[r][kk + kh];
            a0.y = As[r][kk + kh + 1];
            a1.x = As[16 + r][kk + kh];
            a1.y = As[16 + r][kk + kh + 1];
            b0.x = Ws[kk + kh][col0];
            b0.y = Ws[kk + kh + 1][col0];
            b1.x = Ws[kk + kh][col1];
            b1.y = Ws[kk + kh + 1][col1];
            acc00 = __builtin_amdgcn_wmma_f32_16x16x4_f32(
                false, a0, false, b0, (short)0, acc00, false, false);
            acc01 = __builtin_amdgcn_wmma_f32_16x16x4_f32(
                false, a0, false, b1, (short)0, acc01, false, false);
            acc10 = __builtin_amdgcn_wmma_f32_16x16x4_f32(
                false, a1, false, b0, (short)0, acc10, false, false);
            acc11 = __builtin_amdgcn_wmma_f32_16x16x4_f32(
                false, a1, false, b1, (short)0, acc11, false, false);
        }
        __syncthreads();
    }

    // --- store 2x2 tiles; guard N (66*16 = 1056 exactly, so tile-granular) ---
    const int rbase0 = m0 + ((lane >> 4) << 3);
    const int rbase1 = rbase0 + 16;
    const int nc0 = n0 + col0;
    const int nc1 = n0 + col1;
#pragma unroll
    for (int v = 0; v < 8; ++v) {
        if (nc0 < NPROJ) {
            P[(rbase0 + v) * NPROJ + nc0] = acc00[v];
            P[(rbase1 + v) * NPROJ + nc0] = acc10[v];
        }
        if (nc1 < NPROJ) {
            P[(rbase0 + v) * NPROJ + nc1] = acc01[v];
            P[(rbase1 + v) * NPROJ + nc1] = acc11[v];
        }
    }
}

// ---------------------------------------------------------------------------
// Kernel 2: per-(b,d,chunk) affine summary:  h_out = P*h_in + Q over chunk.
// grid (DM/256, NC, BB), block 256 threads over d. B_in staged in LDS.
// ---------------------------------------------------------------------------
__global__ __launch_bounds__(256) void chunk_summary_kernel(
    const float* __restrict__ proj, const float* __restrict__ X,
    const float* __restrict__ A_log,
    float* __restrict__ Pc, float* __restrict__ Qc) {
    __shared__ float sB[CHT][DS];   // 4 KB

    const int t = threadIdx.x;
    const int d = blockIdx.x * 256 + t;
    const int chunk = blockIdx.y;
    const int b = blockIdx.z;
    const int t0 = chunk * CHT;

    // stage B_in: CHT*DS = 1024 floats; one float4 per thread
    {
        int tt = t >> 2;
        int s4 = (t & 3) << 2;
        *(float4*)&sB[tt][s4] =
            *(const float4*)&proj[(size_t)(b * TT + t0 + tt) * NPROJ + DM + s4];
    }
    __syncthreads();

    float negA[DS];
#pragma unroll
    for (int s = 0; s < DS; ++s) negA[s] = -__expf(A_log[d * DS + s]);

    float Pr[DS], Qr[DS];
#pragma unroll
    for (int s = 0; s < DS; ++s) { Pr[s] = 1.0f; Qr[s] = 0.0f; }

    for (int tt = 0; tt < CHT; ++tt) {
        const size_t row = (size_t)(b * TT + t0 + tt);
        const float dr = proj[row * NPROJ + d];
        const float xv = X[row * DM + d];
        const float delta = softplus_f(dr);
        const float dx = delta * xv;
#pragma unroll
        for (int s = 0; s < DS; ++s) {
            const float a = __expf(delta * negA[s]);
            Pr[s] *= a;
            Qr[s] = a * Qr[s] + dx * sB[tt][s];
        }
    }

    const size_t base = (((size_t)(b * DM + d)) * NC + chunk) * DS;
#pragma unroll
    for (int s = 0; s < DS; ++s) {
        Pc[base + s] = Pr[s];
        Qc[base + s] = Qr[s];
    }
}

// ---------------------------------------------------------------------------
// Kernel 3: sequential scan over the 32 chunk summaries per channel.
// Produces each chunk's initial h and the final h_new output.
// ---------------------------------------------------------------------------
__global__ __launch_bounds__(256) void chunk_scan_kernel(
    const float* __restrict__ h0,
    const float* __restrict__ Pc, const float* __restrict__ Qc,
    float* __restrict__ hInit, float* __restrict__ h_new) {
    const int g = blockIdx.x * 256 + threadIdx.x;   // 0..2047 = b*DM + d

    float h[DS];
#pragma unroll
    for (int s = 0; s < DS; ++s) h[s] = h0[(size_t)g * DS + s];

    for (int c = 0; c < NC; ++c) {
        const size_t base = (((size_t)g) * NC + c) * DS;
#pragma unroll
        for (int s = 0; s < DS; ++s) {
            hInit[base + s] = h[s];
            h[s] = Pc[base + s] * h[s] + Qc[base + s];
        }
    }
#pragma unroll
    for (int s = 0; s < DS; ++s) h_new[(size_t)g * DS + s] = h[s];
}

// ---------------------------------------------------------------------------
// Kernel 4: replay each chunk from its initial h and emit y.
// grid (DM/256, NC, BB); B_in and C_in staged in LDS.
// ---------------------------------------------------------------------------
__global__ __launch_bounds__(256) void chunk_output_kernel(
    const float* __restrict__ proj, const float* __restrict__ X,
    const float* __restrict__ A_log, const float* __restrict__ Dv,
    const float* __restrict__ hInit, float* __restrict__ Y) {
    __shared__ float sB[CHT][DS];
    __shared__ float sC[CHT][DS];

    const int t = threadIdx.x;
    const int d = blockIdx.x * 256 + t;
    const int chunk = blockIdx.y;
    const int b = blockIdx.z;
    const int t0 = chunk * CHT;

    {
        int tt = t >> 2;
        int s4 = (t & 3) << 2;
        const size_t row = (size_t)(b * TT + t0 + tt) * NPROJ;
        *(float4*)&sB[tt][s4] = *(const float4*)&proj[row + DM + s4];
        *(float4*)&sC[tt][s4] = *(const float4*)&proj[row + DM + DS + s4];
    }
    __syncthreads();

    float negA[DS];
#pragma unroll
    for (int s = 0; s < DS; ++s) negA[s] = -__expf(A_log[d * DS + s]);

    float h[DS];
    {
        const size_t base = (((size_t)(b * DM + d)) * NC + chunk) * DS;
#pragma unroll
        for (int s = 0; s < DS; ++s) h[s] = hInit[base + s];
    }
    const float Dd = Dv[d];

    for (int tt = 0; tt < CHT; ++tt) {
        const size_t row = (size_t)(b * TT + t0 + tt);
        const float dr = proj[row * NPROJ + d];
        const float xv = X[row * DM + d];
        const float delta = softplus_f(dr);
        const float dx = delta * xv;
        float y = 0.0f;
#pragma unroll
        for (int s = 0; s < DS; ++s) {
            const float a = __expf(delta * negA[s]);
            h[s] = a * h[s] + dx * sB[tt][s];
            y += h[s] * sC[tt][s];
        }
        Y[row * DM + d] = y + Dd * xv;
    }
}

// ---------------------------------------------------------------------------
extern "C" void kernel_launch(void* const* d_in, const int* in_sizes, int n_in,
                              void* d_out, int out_size, void* d_ws, size_t ws_size,
                              hipStream_t stream) {
    const float* x     = (const float*)d_in[0];
    const float* h0    = (const float*)d_in[1];
    const float* W     = (const float*)d_in[2];
    const float* A_log = (const float*)d_in[3];
    const float* Dv    = (const float*)d_in[4];

    float* y     = (float*)d_out;               // (B,T,DM)
    float* h_new = (float*)d_out + Y_ELEMS;     // (B,DM,DS)

    float* proj  = (float*)d_ws;
    float* Pc    = proj + PROJ_ELEMS;
    float* Qc    = Pc + PQ_ELEMS;
    float* hInit = Qc + PQ_ELEMS;

    // 1) projection GEMM (fp32 WMMA, 2x2 tiles/wave)
    gemm_proj_kernel<<<dim3(MM / BM, (NPROJ + BN - 1) / BN), 256, 0, stream>>>(
        x, W, proj);
    // 2) per-chunk affine summaries (P,Q)
    chunk_summary_kernel<<<dim3(DM / 256, NC, BB), 256, 0, stream>>>(
        proj, x, A_log, Pc, Qc);
    // 3) scan over chunks -> per-chunk initial h + final h_new
    chunk_scan_kernel<<<dim3((BB * DM) / 256), 256, 0, stream>>>(
        h0, Pc, Qc, hInit, h_new);
    // 4) replay chunks, emit y
    chunk_output_kernel<<<dim3(DM / 256, NC, BB), 256, 0, stream>>>(
        proj, x, A_log, Dv, hInit, y);
}